// ManifoldNetSPD_231928234155
// MI455X (gfx1250) — compile-verified
//
#include <hip/hip_runtime.h>

// ManifoldNetSPD identity-pad: [8,16,256,256,3,3] -> [8,16,258,258,3,3], p=1.
// Pure data movement (~609 MB total), HBM-bound: ~26us floor @ 23.3 TB/s.
// No FLOPs => no WMMA; we use the CDNA5 async global<->LDS data-mover path
// (ASYNCcnt) for the bulk interior copy, which is the gfx1250-specific
// mechanism for streaming copies.

#define BC          128                     // B*C = 8*16
#define H           256
#define W           256
#define HO          258
#define WO          258
#define CELL        9                       // 3x3 block
#define ROW_FLOATS  (W * CELL)              // 2304 floats = 9216 B per interior row
#define ROW_CHUNKS  (ROW_FLOATS / 4)        // 576 x 16-byte chunks
#define ITHREADS    288                     // 9 wave32; 576/288 = 2 chunks/lane exactly

// ---------------------------------------------------------------------------
// Interior copy: one workgroup per (bc, i) input row.
//   in  row: x  [bc, i,   0:256, :, :]  -> 2304 contiguous floats, 16B-aligned
//   out row: out[bc, i+1, 1:257, :, :]  -> 2304 contiguous floats, 4B-aligned
// Stage through LDS with async b128 transfers (fire-and-forget, ASYNCcnt).
// Output rows are only dword-aligned; DWORD alignment mode permits multi-dword
// global ops at dword alignment (2 LSBs ignored), so b128 stores are legal.
// ---------------------------------------------------------------------------
__global__ __launch_bounds__(ITHREADS)
void spd_pad_interior(const float* __restrict__ x, float* __restrict__ out) {
    const int row = blockIdx.x;             // 0 .. BC*H-1
    const int bc  = row >> 8;               // row / 256
    const int i   = row & 255;              // row % 256

    const float* in_row  = x + (size_t)row * ROW_FLOATS;
    float*       out_row = out + ((size_t)(bc * HO + (i + 1)) * WO + 1) * CELL;

    extern __shared__ float lds_buf[];      // ROW_FLOATS floats at LDS offset 0
    (void)lds_buf;

    const unsigned tid = threadIdx.x;

    // Phase 1: global -> LDS, 16 B per lane per issue (async, ASYNCcnt++)
    #pragma unroll
    for (unsigned k = 0; k < ROW_CHUNKS / ITHREADS; ++k) {
        unsigned byte = (tid + k * ITHREADS) * 16u;
        asm volatile("global_load_async_to_lds_b128 %0, %1, %2"
                     :: "v"(byte), "v"(byte), "s"(in_row)
                     : "memory");
    }
    // Our wave's async loads landed; barrier covers the other 8 waves' chunks.
    asm volatile("s_wait_asynccnt 0" ::: "memory");
    __syncthreads();

    // Phase 2: LDS -> global (async store from LDS)
    #pragma unroll
    for (unsigned k = 0; k < ROW_CHUNKS / ITHREADS; ++k) {
        unsigned byte = (tid + k * ITHREADS) * 16u;
        asm volatile("global_store_async_from_lds_b128 %0, %1, %2"
                     :: "v"(byte), "v"(byte), "s"(out_row)
                     : "memory");
    }
    asm volatile("s_wait_asynccnt 0" ::: "memory");
    // S_ENDPGM performs an implicit wait-idle as well.
}

// ---------------------------------------------------------------------------
// Border fill: 1028 border cells per (bc) image (top/bottom full rows + left/
// right columns of the middle rows); each gets eye(3). 128*1028 cells total,
// ~4.7 MB of stores -- negligible next to the interior stream.
// ---------------------------------------------------------------------------
__global__ __launch_bounds__(256)
void spd_pad_border(float* __restrict__ out) {
    const int per_img = 2 * WO + 2 * H;     // 258 + 258 + 256 + 256 = 1028
    int t = blockIdx.x * blockDim.x + threadIdx.x;
    if (t >= BC * per_img) return;

    int bc = t / per_img;
    int c  = t % per_img;
    int i, j;
    if (c < WO)                { i = 0;                  j = c; }
    else if (c < 2 * WO)       { i = HO - 1;             j = c - WO; }
    else if (c < 2 * WO + H)   { i = c - 2 * WO + 1;     j = 0; }
    else                       { i = c - (2 * WO + H) + 1; j = WO - 1; }

    float* cell = out + ((size_t)(bc * HO + i) * WO + j) * CELL;
    cell[0] = 1.0f; cell[1] = 0.0f; cell[2] = 0.0f;
    cell[3] = 0.0f; cell[4] = 1.0f; cell[5] = 0.0f;
    cell[6] = 0.0f; cell[7] = 0.0f; cell[8] = 1.0f;
}

// ---------------------------------------------------------------------------
extern "C" void kernel_launch(void* const* d_in, const int* in_sizes, int n_in,
                              void* d_out, int out_size, void* d_ws, size_t ws_size,
                              hipStream_t stream) {
    (void)in_sizes; (void)n_in; (void)out_size; (void)d_ws; (void)ws_size;
    const float* x   = (const float*)d_in[0];   // [8,16,256,256,3,3] f32
    float*       out = (float*)d_out;           // [8,16,258,258,3,3] f32
    // d_in[1] is padding_dim == 1 (shapes hardcoded to the reference).

    // Interior: 32768 rows, one workgroup each, 9216 B dynamic LDS.
    spd_pad_interior<<<BC * H, ITHREADS, ROW_FLOATS * sizeof(float), stream>>>(x, out);

    // Border: disjoint from interior writes, so order vs. interior is irrelevant.
    int ncells = BC * (2 * WO + 2 * H);
    spd_pad_border<<<(ncells + 255) / 256, 256, 0, stream>>>(out);
}